// HypergraphNN_35716948033729
// MI455X (gfx1250) — compile-verified
//
#include <hip/hip_runtime.h>
#include <hip/hip_bf16.h>

typedef float v2f __attribute__((ext_vector_type(2)));
typedef float v8f __attribute__((ext_vector_type(8)));

static constexpr int DIM = 8192;   // N_NODES == N_EDGES
static constexpr int FD  = 128;    // FEAT_DIM
static constexpr int BM  = 32;     // output rows per block
static constexpr int BK  = 32;     // K chunk
static constexpr float EPSV = 1e-10f;

// ---------------------------------------------------------------------------
// Degree kernels
// ---------------------------------------------------------------------------

// One block per row: dvis[row] = 1/sqrt(rowsum + eps)
__global__ __launch_bounds__(256) void HG_rowdeg(const float* __restrict__ H,
                                                 float* __restrict__ dvis) {
    __shared__ float red[256];
    const int row = blockIdx.x;
    const float* p = H + (size_t)row * DIM;
    float s = 0.0f;
    for (int c = threadIdx.x; c < DIM; c += 256) s += p[c];
    red[threadIdx.x] = s;
    __syncthreads();
    for (int off = 128; off > 0; off >>= 1) {
        if (threadIdx.x < off) red[threadIdx.x] += red[threadIdx.x + off];
        __syncthreads();
    }
    if (threadIdx.x == 0) dvis[row] = 1.0f / sqrtf(red[0] + EPSV);
}

__global__ __launch_bounds__(256) void HG_colzero(float* __restrict__ de) {
    de[blockIdx.x * 256 + threadIdx.x] = 0.0f;
}

// grid (32, 32): block (bx,by) sums rows [by*256, by*256+256) of cols bx*256..
__global__ __launch_bounds__(256) void HG_coldeg(const float* __restrict__ H,
                                                 float* __restrict__ de) {
    const int c  = blockIdx.x * 256 + threadIdx.x;
    const int r0 = blockIdx.y * 256;
    float s = 0.0f;
    #pragma unroll 4
    for (int r = 0; r < 256; ++r) s += H[(size_t)(r0 + r) * DIM + c];
    atomicAdd(&de[c], s);
}

__global__ __launch_bounds__(256) void HG_colfin(float* __restrict__ de) {
    const int i = blockIdx.x * 256 + threadIdx.x;
    de[i] = 1.0f / (de[i] + EPSV);
}

// ---------------------------------------------------------------------------
// WMMA f32 GEMM, templated on TRANSA / OUTSCALE, double-buffered LDS pipeline:
//   out[m, d] = outScale[m] * sum_k A'(m,k) * (scaleB[k] * srcB[k, d])
//   A'(m,k) = TRANSA ? A[k*DIM + m] : A[m*DIM + k]
// Block: 256 threads = 8 wave32s; 32 rows x 128 cols of output.
// Wave w owns column tile [16w, 16w+16) and two 16-row subtiles.
// Both A and B are staged in LDS as interleaved k-pairs so every WMMA
// fragment is a single b64 LDS read into an adjacent even-aligned VGPR pair.
// ---------------------------------------------------------------------------
template <bool TRANSA, bool OUTSCALE>
__global__ __launch_bounds__(256) void HG_gemm(const float* __restrict__ A,
                                               const float* __restrict__ srcB,
                                               const float* __restrict__ scaleB,
                                               const float* __restrict__ outScale,
                                               float* __restrict__ out) {
    // lA pair layout: lA2[kp*BM + m]  = {A'(m,2kp),  A'(m,2kp+1)}
    // lB pair layout: lB2[kp*FD + d]  = {Bs(2kp,d),  Bs(2kp+1,d)},  Bs = scaleB*srcB
    __shared__ __align__(16) float lAf[2][(BK / 2) * BM * 2];   // 2 x 4 KB
    __shared__ __align__(16) float lBf[2][(BK / 2) * FD * 2];   // 2 x 16 KB

    const int tid  = threadIdx.x;
    const int wave = tid >> 5;        // 0..7 -> column tile
    const int lid  = tid & 31;
    const int half = lid >> 4;        // 0: lanes 0-15, 1: lanes 16-31
    const int ln   = lid & 15;
    const int m0   = blockIdx.x * BM;

    // per-thread staging roles (compile-time layout, no divergence)
    const int a_i4 = (tid & 7) * 4;   // TRANSA: m quad ; else: k quad
    const int a_j  = tid >> 3;        // TRANSA: k      ; else: m
    const int b_d4 = (tid & 31) * 4;
    const int b_k  = tid >> 5;        // +8 per pass

    v8f acc[2];
    acc[0] = (v8f){};
    acc[1] = (v8f){};

    float4 ra;          // staged A piece for next chunk
    float4 rb[4];       // staged B pieces for next chunk
    float  sb[4];       // B row scales for next chunk

    auto load_chunk = [&](int k0) {
        if (TRANSA) {
            // A'(m,k) = A[(k0+k)*DIM + m0+m]; float4 along m
            ra = *(const float4*)(A + (size_t)(k0 + a_j) * DIM + m0 + a_i4);
        } else {
            // A'(m,k) = A[(m0+m)*DIM + k0+k]; float4 along k
            ra = *(const float4*)(A + (size_t)(m0 + a_j) * DIM + k0 + a_i4);
        }
        #pragma unroll
        for (int p = 0; p < 4; ++p) {
            const int row = k0 + b_k + p * 8;
            sb[p] = scaleB[row];
            rb[p] = *(const float4*)(srcB + (size_t)row * FD + b_d4);
        }
    };

    auto store_chunk = [&](int buf) {
        if (TRANSA) {
            const int kp = a_j >> 1, kc = a_j & 1;
            lAf[buf][(kp * BM + a_i4 + 0) * 2 + kc] = ra.x;
            lAf[buf][(kp * BM + a_i4 + 1) * 2 + kc] = ra.y;
            lAf[buf][(kp * BM + a_i4 + 2) * 2 + kc] = ra.z;
            lAf[buf][(kp * BM + a_i4 + 3) * 2 + kc] = ra.w;
        } else {
            lAf[buf][((a_i4 >> 1) * BM + a_j) * 2 + 0] = ra.x;
            lAf[buf][((a_i4 >> 1) * BM + a_j) * 2 + 1] = ra.y;
            lAf[buf][(((a_i4 + 2) >> 1) * BM + a_j) * 2 + 0] = ra.z;
            lAf[buf][(((a_i4 + 2) >> 1) * BM + a_j) * 2 + 1] = ra.w;
        }
        #pragma unroll
        for (int p = 0; p < 4; ++p) {
            const int k  = b_k + p * 8;
            const int kp = k >> 1, kc = k & 1;
            lBf[buf][(kp * FD + b_d4 + 0) * 2 + kc] = rb[p].x * sb[p];
            lBf[buf][(kp * FD + b_d4 + 1) * 2 + kc] = rb[p].y * sb[p];
            lBf[buf][(kp * FD + b_d4 + 2) * 2 + kc] = rb[p].z * sb[p];
            lBf[buf][(kp * FD + b_d4 + 3) * 2 + kc] = rb[p].w * sb[p];
        }
    };

    // ---- pipeline prologue: chunk 0 into buffer 0
    load_chunk(0);
    store_chunk(0);
    __syncthreads();

    int cur = 0;
    for (int ki = 0; ki < DIM / BK; ++ki) {
        // issue global loads for the next chunk (wraps on last iter; unused)
        const int knext = ((ki + 1) & (DIM / BK - 1)) * BK;
        load_chunk(knext);

        // ---- 8 WMMA steps of K=4 each on lds[cur]
        const float2* lA2 = (const float2*)lAf[cur];
        const float2* lB2 = (const float2*)lBf[cur];
        #pragma unroll
        for (int s = 0; s < 8; ++s) {
            // pair index: lanes 0-15 take K = {4s,4s+1}, lanes 16-31 {4s+2,4s+3}
            const int kp = 2 * s + half;
            const float2 bf  = lB2[kp * FD + wave * 16 + ln];
            const float2 a0f = lA2[kp * BM + ln];
            const float2 a1f = lA2[kp * BM + 16 + ln];
            v2f b;  b[0]  = bf.x;  b[1]  = bf.y;
            v2f a0; a0[0] = a0f.x; a0[1] = a0f.y;
            v2f a1; a1[0] = a1f.x; a1[1] = a1f.y;
            acc[0] = __builtin_amdgcn_wmma_f32_16x16x4_f32(
                false, a0, false, b, (short)0, acc[0], false, false);
            acc[1] = __builtin_amdgcn_wmma_f32_16x16x4_f32(
                false, a1, false, b, (short)0, acc[1], false, false);
        }

        // ---- stage next chunk into the other buffer, one barrier per iter
        store_chunk(cur ^ 1);
        __syncthreads();
        cur ^= 1;
    }

    // ---- store: C/D layout — VGPR j: lanes0-15 M=j, lanes16-31 M=j+8; N=ln
    #pragma unroll
    for (int r = 0; r < 2; ++r) {
        #pragma unroll
        for (int j = 0; j < 8; ++j) {
            const int row = m0 + 16 * r + half * 8 + j;
            const float os = OUTSCALE ? outScale[row] : 1.0f;
            out[(size_t)row * FD + wave * 16 + ln] = acc[r][j] * os;
        }
    }
}

// ---------------------------------------------------------------------------
extern "C" void kernel_launch(void* const* d_in, const int* in_sizes, int n_in,
                              void* d_out, int out_size, void* d_ws, size_t ws_size,
                              hipStream_t stream) {
    (void)in_sizes; (void)n_in; (void)out_size; (void)ws_size;
    const float* H = (const float*)d_in[0];   // [8192, 8192]
    const float* F = (const float*)d_in[1];   // [8192, 128]
    float* out  = (float*)d_out;              // [8192, 128]
    float* ws   = (float*)d_ws;
    float* dvis = ws;                 // 8192 floats: 1/sqrt(d_v + eps)
    float* de   = ws + DIM;           // 8192 floats: 1/(d_e + eps)
    float* Y    = ws + 2 * DIM;       // 8192*128 floats: intermediate H^T X

    HG_rowdeg<<<DIM, 256, 0, stream>>>(H, dvis);
    HG_colzero<<<DIM / 256, 256, 0, stream>>>(de);
    HG_coldeg<<<dim3(DIM / 256, DIM / 256), 256, 0, stream>>>(H, de);
    HG_colfin<<<DIM / 256, 256, 0, stream>>>(de);

    // Y = H^T (F * dvis[row])        : TRANSA, no output scale
    HG_gemm<true, false><<<DIM / BM, 256, 0, stream>>>(H, F, dvis, nullptr, Y);
    // out = dvis[row] * ( H (Y * de[row]) )
    HG_gemm<false, true><<<DIM / BM, 256, 0, stream>>>(H, Y, de, dvis, out);
}